// SparseEdgeKoopman_57097295233746
// MI455X (gfx1250) — compile-verified
//
#include <hip/hip_runtime.h>
#include <math.h>

// ---------------- problem constants ----------------
constexpr int Bdim = 4, T = 256, N = 128, F = 64, H = 128, L = 32, Medge = 64;
constexpr int K_TOP = 16, K_ADJ = 1638, W = 10;
constexpr float EPS = 1e-8f;

// padded LDS leading dims (bank-conflict-free fragment gathers)
constexpr int PXF = F + 1;   // 65
constexpr int PH  = H + 1;   // 129
constexpr int PL  = L + 1;   // 33
constexpr int PN  = N + 1;   // 129

// dynamic LDS layout (floats)
constexpr int OFF_BUF0 = 0;              // 128*PN = 16512 : H1 / D1 / A
constexpr int OFF_BUF1 = 16512;          // 16512 : X/Xn (first 8320) then G
constexpr int OFF_BUF2 = 33024;          // 4224  : Z / Zn
constexpr int OFF_SCR  = 37248;          // 1024  : hist / scan / reduce scratch
constexpr int SMEM_FLOATS = 38272;       // ~150 KB of the WGP's 320 KB LDS

// output layout: X_recon, Z, recon_loss, adj_loss, loss_rss
constexpr size_t OUT_XREC  = 0;
constexpr size_t OUT_Z     = (size_t)Bdim * T * N * F;          // 8388608
constexpr size_t OUT_RECON = OUT_Z + (size_t)Bdim * T * N * L;  // 12582912
constexpr size_t OUT_ADJ   = OUT_RECON + 1;
constexpr size_t OUT_RSS   = OUT_RECON + 2;

typedef float v2f __attribute__((ext_vector_type(2)));
typedef float v8f __attribute__((ext_vector_type(8)));

// order-preserving float->uint key (larger key == larger float)
__device__ inline unsigned fkey(float x) {
  unsigned u = __float_as_uint(x);
  return (u & 0x80000000u) ? ~u : (u | 0x80000000u);
}

// One 16x16 f32 output tile via V_WMMA_F32_16X16X4_F32, accumulating over K.
// A: row-major MxK (LDS). B: row-major KxN if !BT, else row-major NxK (i.e. B^T).
template <bool BT>
__device__ inline v8f gemm_tile(const float* __restrict__ A, int lda,
                                const float* __restrict__ Bm, int ldb,
                                int r0, int c0, int K, int lane, v8f acc) {
  const int hf = lane >> 4, lid = lane & 15;
#pragma unroll 4
  for (int k0 = 0; k0 < K; k0 += 4) {
    v2f a, b;
    // A frag 16x4: lanes 0-15 K={k0,k0+1}, lanes 16-31 K={k0+2,k0+3}
    a.x = A[(r0 + lid) * lda + (k0 + 2 * hf + 0)];
    a.y = A[(r0 + lid) * lda + (k0 + 2 * hf + 1)];
    if constexpr (BT) {
      b.x = Bm[(c0 + lid) * ldb + (k0 + 2 * hf + 0)];
      b.y = Bm[(c0 + lid) * ldb + (k0 + 2 * hf + 1)];
    } else {
      b.x = Bm[(k0 + 2 * hf + 0) * ldb + (c0 + lid)];
      b.y = Bm[(k0 + 2 * hf + 1) * ldb + (c0 + lid)];
    }
    acc = __builtin_amdgcn_wmma_f32_16x16x4_f32(false, a, false, b, (short)0,
                                                acc, false, false);
  }
  return acc;
}

// ================= kernel 0: zero the scalar accumulators =================
__global__ void zero_scalars(float* out) {
  if ((int)threadIdx.x < 3) out[OUT_RECON + threadIdx.x] = 0.f;
}

// ================= kernel 1: fused per-(b,t) pipeline =================
__global__ __launch_bounds__(256) void fused_bt_kernel(
    const float* __restrict__ X, const int* __restrict__ nem,
    const float* __restrict__ We1, const float* __restrict__ be1,
    const float* __restrict__ We2, const float* __restrict__ be2,
    const float* __restrict__ Wd1, const float* __restrict__ bd1,
    const float* __restrict__ Wd2, const float* __restrict__ bd2,
    float* __restrict__ out, float* __restrict__ Gt) {
  extern __shared__ float smem[];
  float* sH = smem + OFF_BUF0;  // 128 x PH : H1 then D1, later A (128 x PN)
  float* sA = smem + OFF_BUF0;
  float* sX = smem + OFF_BUF1;  // 128 x PXF : X then Xn; region later holds G
  float* sG = smem + OFF_BUF1;  // 128 x PN
  float* sZ = smem + OFF_BUF2;  // 128 x PL : Z then Zn
  float* scr = smem + OFF_SCR;

  const int tid = threadIdx.x;
  const int wv = tid >> 5, lane = tid & 31;
  const int hf = lane >> 4, lid = lane & 15;
  const int bt = blockIdx.x;
  const int b = bt / T, t = bt % T;

  const float* Xg = X + (size_t)bt * (N * F);

  // ---- load X tile ----
  for (int i = tid; i < N * F; i += 256) sX[(i >> 6) * PXF + (i & 63)] = Xg[i];
  __syncthreads();

  // ---- H1 = relu(X @ We1 + be1), 128x128 K=64 ----
  {
    int r0 = wv * 16;
    for (int ct = 0; ct < 8; ++ct) {
      int c0 = ct * 16;
      v8f acc = {};
      acc = gemm_tile<false>(sX, PXF, We1, H, r0, c0, F, lane, acc);
      float bv = be1[c0 + lid];
#pragma unroll
      for (int v = 0; v < 8; ++v) {
        float x = acc[v] + bv;
        sH[(r0 + v + 8 * hf) * PH + c0 + lid] = x > 0.f ? x : 0.f;
      }
    }
  }
  __syncthreads();

  // ---- Z = H1 @ We2 + be2, 128x32 K=128 (store LDS + global) ----
  {
    int r0 = wv * 16;
    float* Zout = out + OUT_Z + (size_t)bt * (N * L);
    for (int ct = 0; ct < 2; ++ct) {
      int c0 = ct * 16;
      v8f acc = {};
      acc = gemm_tile<false>(sH, PH, We2, L, r0, c0, H, lane, acc);
      float bv = be2[c0 + lid];
#pragma unroll
      for (int v = 0; v < 8; ++v) {
        float z = acc[v] + bv;
        int rr = r0 + v + 8 * hf;
        sZ[rr * PL + c0 + lid] = z;
        Zout[rr * L + c0 + lid] = z;
      }
    }
  }
  __syncthreads();

  // ---- D1 = relu(Z @ Wd1 + bd1), 128x128 K=32 (overwrite sH) ----
  {
    int r0 = wv * 16;
    for (int ct = 0; ct < 8; ++ct) {
      int c0 = ct * 16;
      v8f acc = {};
      acc = gemm_tile<false>(sZ, PL, Wd1, H, r0, c0, L, lane, acc);
      float bv = bd1[c0 + lid];
#pragma unroll
      for (int v = 0; v < 8; ++v) {
        float x = acc[v] + bv;
        sH[(r0 + v + 8 * hf) * PH + c0 + lid] = x > 0.f ? x : 0.f;
      }
    }
  }
  __syncthreads();

  // ---- X_recon = D1 @ Wd2 + bd2, 128x64 K=128; recon loss partial ----
  float rec_sum = 0.f;
  {
    int r0 = wv * 16;
    float* Rout = out + OUT_XREC + (size_t)bt * (N * F);
    for (int ct = 0; ct < 4; ++ct) {
      int c0 = ct * 16;
      v8f acc = {};
      acc = gemm_tile<false>(sH, PH, Wd2, F, r0, c0, H, lane, acc);
      float bv = bd2[c0 + lid];
#pragma unroll
      for (int v = 0; v < 8; ++v) {
        int rr = r0 + v + 8 * hf;
        float xr = acc[v] + bv;
        Rout[rr * F + c0 + lid] = xr;
        float d = sX[rr * PXF + c0 + lid] - xr;
        rec_sum += d * d;
      }
    }
  }
  __syncthreads();

  // ---- normalize X rows in place -> Xn ----
  if (tid < N) {
    float* row = sX + tid * PXF;
    float m = 0.f;
    for (int k = 0; k < F; ++k) m += row[k];
    m *= (1.0f / F);
    float s = 0.f;
    for (int k = 0; k < F; ++k) { float c = row[k] - m; s += c * c; }
    float inv = 1.0f / (sqrtf(s) + EPS);
    for (int k = 0; k < F; ++k) row[k] = (row[k] - m) * inv;
  }
  __syncthreads();

  // ---- A = Xn @ Xn^T, 128x128 K=64 (into buf0) ----
  {
    int r0 = wv * 16;
    for (int ct = 0; ct < 8; ++ct) {
      int c0 = ct * 16;
      v8f acc = {};
      acc = gemm_tile<true>(sX, PXF, sX, PXF, r0, c0, F, lane, acc);
#pragma unroll
      for (int v = 0; v < 8; ++v)
        sA[(r0 + v + 8 * hf) * PN + c0 + lid] = acc[v];
    }
  }
  __syncthreads();

  // ---- normalize Z rows in place -> Zn ----
  if (tid < N) {
    float* row = sZ + tid * PL;
    float m = 0.f;
    for (int k = 0; k < L; ++k) m += row[k];
    m *= (1.0f / L);
    float s = 0.f;
    for (int k = 0; k < L; ++k) { float c = row[k] - m; s += c * c; }
    float inv = 1.0f / (sqrtf(s) + EPS);
    for (int k = 0; k < L; ++k) row[k] = (row[k] - m) * inv;
  }
  __syncthreads();

  // ---- G = Zn @ Zn^T, 128x128 K=32 (into buf1, overwrites Xn) ----
  {
    int r0 = wv * 16;
    for (int ct = 0; ct < 8; ++ct) {
      int c0 = ct * 16;
      v8f acc = {};
      acc = gemm_tile<true>(sZ, PL, sZ, PL, r0, c0, L, lane, acc);
#pragma unroll
      for (int v = 0; v < 8; ++v)
        sG[(r0 + v + 8 * hf) * PN + c0 + lid] = acc[v];
    }
  }
  __syncthreads();

  // ---- per-row top-16 of A_u (64 candidates via nem), gather G -> Gt ----
  // wave wv handles rows wv*16 .. wv*16+15; jax tie-break: lowest index wins.
  for (int ri = 0; ri < 16; ++ri) {
    int row = wv * 16 + ri;
    const int* nrow = nem + row * Medge;
    int e0 = nrow[lane], e1 = nrow[lane + 32];
    float v0 = sA[(e0 >> 7) * PN + (e0 & 127)];
    float v1 = sA[(e1 >> 7) * PN + (e1 & 127)];
    for (int sel = 0; sel < K_TOP; ++sel) {
      float bv; int bj;
      if (v0 >= v1) { bv = v0; bj = lane; } else { bv = v1; bj = lane + 32; }
      for (int off = 16; off > 0; off >>= 1) {
        float ov = __shfl_xor(bv, off, 32);
        int oj = __shfl_xor(bj, off, 32);
        if (ov > bv || (ov == bv && oj < bj)) { bv = ov; bj = oj; }
      }
      if (lane == sel) {
        int e = nrow[bj];
        Gt[(((size_t)b * N + row) * T + t) * K_TOP + sel] =
            sG[(e >> 7) * PN + (e & 127)];
      }
      if (bj == lane) v0 = -1e30f;
      else if (bj == lane + 32) v1 = -1e30f;
    }
  }
  __syncthreads();

  // ---- adjacency loss: exact top-1638 of A (radix select + index-order ties)
  int* hist = (int*)scr;          // 256
  int* ibc = hist + 256;          // 4
  float* red = (float*)(ibc + 4); // 256
  int* tiescan = (int*)(red + 256); // 256

  unsigned prefix = 0, msk = 0;
  int kRem = K_ADJ;
  for (int shift = 24; shift >= 0; shift -= 8) {
    hist[tid] = 0;
    __syncthreads();
    for (int i = 0; i < 64; ++i) {
      int idx = tid * 64 + i;
      unsigned key = fkey(sA[(idx >> 7) * PN + (idx & 127)]);
      if ((key & msk) == prefix) atomicAdd(&hist[(key >> shift) & 255], 1);
    }
    __syncthreads();
    if (tid == 0) {
      int cum = 0, digit = 0, kr = kRem;
      for (int bin = 255; bin >= 0; --bin) {
        int h = hist[bin];
        if (cum + h >= kr) { digit = bin; break; }
        cum += h;
      }
      ibc[0] = digit;
      ibc[1] = kRem - cum;
    }
    __syncthreads();
    prefix |= ((unsigned)ibc[0]) << shift;
    msk |= 0xFFu << shift;
    kRem = ibc[1];
    __syncthreads();
  }
  const unsigned tau = prefix;
  const int r_ties = kRem;  // #elements equal to tau to include (index order)

  float lsum = 0.f;
  int tieCnt = 0;
  for (int i = 0; i < 64; ++i) {
    int idx = tid * 64 + i;
    float a = sA[(idx >> 7) * PN + (idx & 127)];
    unsigned key = fkey(a);
    if (key > tau) {
      float d = sG[(idx >> 7) * PN + (idx & 127)] - a;
      lsum += d * d;
    } else if (key == tau) {
      ++tieCnt;
    }
  }
  tiescan[tid] = tieCnt;
  __syncthreads();
  for (int off = 1; off < 256; off <<= 1) {
    int v = (tid >= off) ? tiescan[tid - off] : 0;
    __syncthreads();
    tiescan[tid] += v;
    __syncthreads();
  }
  int pre = tiescan[tid] - tieCnt;
  int take = r_ties - pre;
  if (take > tieCnt) take = tieCnt;
  if (take > 0) {
    int got = 0;
    for (int i = 0; i < 64 && got < take; ++i) {
      int idx = tid * 64 + i;
      float a = sA[(idx >> 7) * PN + (idx & 127)];
      if (fkey(a) == tau) {
        float d = sG[(idx >> 7) * PN + (idx & 127)] - a;
        lsum += d * d;
        ++got;
      }
    }
  }

  // ---- block reduce recon + adjacency partials ----
  float wr = rec_sum, wa = lsum;
  for (int off = 16; off > 0; off >>= 1) {
    wr += __shfl_xor(wr, off, 32);
    wa += __shfl_xor(wa, off, 32);
  }
  __syncthreads();
  if (lane == 0) { red[wv] = wr; red[8 + wv] = wa; }
  __syncthreads();
  if (tid == 0) {
    float s1 = 0.f, s2 = 0.f;
    for (int i = 0; i < 8; ++i) { s1 += red[i]; s2 += red[8 + i]; }
    atomicAdd(out + OUT_RECON, s1 * (1.0f / ((float)Bdim * T * N * F)));
    atomicAdd(out + OUT_ADJ, s2 * (1.0f / ((float)Bdim * T * K_ADJ)));
  }
}

// ================= kernel 2: windowed LKIS loss =================
// per (b,n): band Gram trick + per-window 9x9 Cholesky.
// sum|resid|^2 = |Y1|^2 - <C, C G^-1> - eps*|C G^-1|^2  with G = Y0 Y0^T + eps I
__global__ __launch_bounds__(256) void lkis_kernel(const float* __restrict__ Gt,
                                                   float* __restrict__ out) {
  __shared__ float sRow[T * K_TOP];  // 256 x 16
  __shared__ float sBand[T * W];     // dot(row_t, row_{t+d}), d<10
  __shared__ float red[8];

  const int tid = threadIdx.x;
  const int bn = blockIdx.x;
  const float* src = Gt + (size_t)bn * (T * K_TOP);
  for (int i = tid; i < T * K_TOP; i += 256) sRow[i] = src[i];
  __syncthreads();

  if (tid < T) {
    for (int d = 0; d < W; ++d) {
      float s = 0.f;
      if (tid + d < T) {
        const float* r0 = sRow + tid * K_TOP;
        const float* r1 = sRow + (tid + d) * K_TOP;
#pragma unroll
        for (int k = 0; k < K_TOP; ++k) s += r0[k] * r1[k];
      }
      sBand[tid * W + d] = s;
    }
  }
  __syncthreads();

  float lsum = 0.f;
  const int NW = T - W + 1;  // 247
  if (tid < NW) {
    const int w0 = tid;
    float g[9][9], c[9][9];
    float term1 = 0.f;
    for (int i = 0; i < 9; ++i) {
      for (int j = 0; j < 9; ++j) {
        int lo = i < j ? i : j, d = i < j ? j - i : i - j;
        g[i][j] = sBand[(w0 + lo) * W + d] + (i == j ? 1e-6f : 0.f);
        int i1 = i + 1;
        int lo2 = i1 < j ? i1 : j, d2 = i1 < j ? j - i1 : i1 - j;
        c[i][j] = sBand[(w0 + lo2) * W + d2];
      }
      term1 += sBand[(w0 + i + 1) * W + 0];
    }
    // Cholesky (L in lower triangle of g)
    for (int i = 0; i < 9; ++i)
      for (int j = 0; j <= i; ++j) {
        float s = g[i][j];
        for (int k = 0; k < j; ++k) s -= g[i][k] * g[j][k];
        if (i == j) g[i][i] = sqrtf(s);
        else g[i][j] = s / g[j][j];
      }
    float t2 = 0.f, t3 = 0.f;
    for (int i = 0; i < 9; ++i) {
      float y[9];
      for (int p = 0; p < 9; ++p) {
        float s = c[i][p];
        for (int q = 0; q < p; ++q) s -= g[p][q] * y[q];
        y[p] = s / g[p][p];
      }
      for (int p = 8; p >= 0; --p) {
        float s = y[p];
        for (int q = p + 1; q < 9; ++q) s -= g[q][p] * y[q];
        y[p] = s / g[p][p];
      }
      for (int p = 0; p < 9; ++p) { t2 += y[p] * c[i][p]; t3 += y[p] * y[p]; }
    }
    lsum = term1 - t2 - 1e-6f * t3;
  }
  for (int off = 16; off > 0; off >>= 1) lsum += __shfl_xor(lsum, off, 32);
  int wv = tid >> 5, lane = tid & 31;
  if (lane == 0) red[wv] = lsum;
  __syncthreads();
  if (tid == 0) {
    float s = 0.f;
    for (int i = 0; i < 8; ++i) s += red[i];
    // divisor: B*N*nw*9*16 = 4*128*247*144
    atomicAdd(out + OUT_RSS, s * (1.0f / 18210816.0f));
  }
}

// ================= host launcher =================
extern "C" void kernel_launch(void* const* d_in, const int* in_sizes, int n_in,
                              void* d_out, int out_size, void* d_ws,
                              size_t ws_size, hipStream_t stream) {
  (void)in_sizes; (void)n_in; (void)out_size; (void)ws_size;
  const float* X = (const float*)d_in[0];
  const int* nem = (const int*)d_in[1];
  const float* We1 = (const float*)d_in[2];
  const float* be1 = (const float*)d_in[3];
  const float* We2 = (const float*)d_in[4];
  const float* be2 = (const float*)d_in[5];
  const float* Wd1 = (const float*)d_in[6];
  const float* bd1 = (const float*)d_in[7];
  const float* Wd2 = (const float*)d_in[8];
  const float* bd2 = (const float*)d_in[9];
  float* out = (float*)d_out;
  float* Gt = (float*)d_ws;  // B*N*T*K_TOP floats = 8 MB

  const size_t smem = (size_t)SMEM_FLOATS * sizeof(float);  // ~150 KB
  (void)hipFuncSetAttribute((const void*)fused_bt_kernel,
                            hipFuncAttributeMaxDynamicSharedMemorySize,
                            (int)smem);

  zero_scalars<<<1, 32, 0, stream>>>(out);
  fused_bt_kernel<<<dim3(Bdim * T), dim3(256), smem, stream>>>(
      X, nem, We1, be1, We2, be2, Wd1, bd1, Wd2, bd2, out, Gt);
  lkis_kernel<<<dim3(Bdim * N), dim3(256), 0, stream>>>(Gt, out);
}